// LLMEmbeddingVQ_3753801417215
// MI455X (gfx1250) — compile-verified
//
#include <hip/hip_runtime.h>
#include <hip/hip_bf16.h>

typedef __attribute__((ext_vector_type(2))) float v2f;
typedef __attribute__((ext_vector_type(8))) float v8f;

// ---------------- problem constants ----------------
constexpr int  D_IN   = 512;
constexpr int  D_EMB  = 1024;
constexpr int  K_CB   = 4096;
constexpr int  T_TOT  = 4 * 32 * 256;        // 32768 tokens
constexpr int  TOKT   = 32;                  // tokens per block
constexpr int  THREADS= 512;                 // 16 wave32
constexpr int  NWAVES = THREADS / 32;
constexpr int  ZST    = 1026;                // padded LDS row stride (floats), 8B aligned, bank-conflict free

// d_out layout (floats), concatenated in reference return order
constexpr size_t OUT_OFF = 0;                                  // [T_TOT, 512]
constexpr size_t IDX_OFF = (size_t)T_TOT * D_IN;               // [T_TOT] (as float values)
constexpr size_t Q_OFF   = IDX_OFF + (size_t)T_TOT;            // [T_TOT, 1024]
constexpr size_t AUX_OFF = Q_OFF + (size_t)T_TOT * D_EMB;      // scalar

// workspace layout (floats)
constexpr size_t WS_CNORM    = 0;            // K_CB floats
constexpr size_t WS_PARTIALS = K_CB;         // T_TOT/TOKT = 1024 floats
constexpr int    NBLOCKS     = T_TOT / TOKT; // 1024

// LDS: zs + qs + min tables + idx + reduce buffer
constexpr int LDS_FLOATS = TOKT * ZST * 2 + 32 * 8 /*minv*/ + 32 * 8 /*mini*/ + 32 /*idx*/ + THREADS /*red*/;
constexpr size_t LDS_BYTES = (size_t)LDS_FLOATS * 4;   // 266,880 B < 320 KB

#define WMMA_F32(A, B, C) \
  __builtin_amdgcn_wmma_f32_16x16x4_f32(false, (A), false, (B), (short)0, (C), false, false)

// ---------------- kernel 0: codebook squared norms ----------------
__global__ __launch_bounds__(256) void vq_cnorm_kernel(const float* __restrict__ cb,
                                                       float* __restrict__ cnorm) {
  int lane = threadIdx.x & 31;
  int wv   = threadIdx.x >> 5;
  int row  = blockIdx.x * 8 + wv;            // grid = K_CB/8
  const float* r = cb + (size_t)row * D_EMB;
  float s = 0.f;
  #pragma unroll 4
  for (int e = lane; e < D_EMB; e += 32) { float v = r[e]; s += v * v; }
  #pragma unroll
  for (int m = 16; m >= 1; m >>= 1) s += __shfl_xor(s, m, 32);
  if (lane == 0) cnorm[row] = s;
}

// ---------------- kernel 1: fused VQ pipeline ----------------
__global__ __launch_bounds__(THREADS, 1)
void vq_fused_kernel(const float* __restrict__ x,     // [T_TOT, 512]
                     const float* __restrict__ w_in,  // [1024, 512]
                     const float* __restrict__ b_in,  // [1024]
                     const float* __restrict__ w_out, // [512, 1024]
                     const float* __restrict__ b_out, // [512]
                     const float* __restrict__ cb,    // [4096, 1024]
                     const float* __restrict__ cnorm, // [4096]
                     float* __restrict__ partials,    // [NBLOCKS]
                     float* __restrict__ outf) {      // d_out (float view)
  extern __shared__ char smem_raw[];
  float* zs   = (float*)smem_raw;                 // [TOKT][ZST]
  float* qs   = zs + TOKT * ZST;                  // [TOKT][ZST]
  float* minv = qs + TOKT * ZST;                  // [32 rows][8 wave-slots]
  int*   mini = (int*)(minv + 32 * 8);            // [32][8]
  int*   idxs = mini + 32 * 8;                    // [32]
  float* red  = (float*)(idxs + 32);              // [THREADS]

  const int tid  = threadIdx.x;
  const int lane = tid & 31;
  const int wave = tid >> 5;
  const int half = lane >> 4;        // 0: K={0,1}, 1: K={2,3}
  const int l15  = lane & 15;
  const int kb   = half * 2;
  const int tok0 = blockIdx.x * TOKT;

  // ---------- phase 1: z = x @ w_in^T + b_in  -> LDS ----------
  // 2 token-subtiles x 32 column-pairs = 64 units; A fragment shared by the pair.
  for (int unit = wave; unit < 64; unit += NWAVES) {
    const int msub = unit >> 5;
    const int n0   = (unit & 31) << 5;          // 32-wide column pair
    v8f acc0 = {}, acc1 = {};
    const float* xr  = x    + (size_t)(tok0 + msub * 16 + l15) * D_IN + kb;
    const float* wr0 = w_in + (size_t)(n0 + l15) * D_IN + kb;
    const float* wr1 = wr0  + (size_t)16 * D_IN;
    #pragma unroll 4
    for (int kk = 0; kk < D_IN; kk += 4) {
      v2f a  = *(const v2f*)(xr + kk);
      v2f b0 = *(const v2f*)(wr0 + kk);
      v2f b1 = *(const v2f*)(wr1 + kk);
      acc0 = WMMA_F32(a, b0, acc0);
      acc1 = WMMA_F32(a, b1, acc1);
    }
    const float bias0 = b_in[n0 + l15];
    const float bias1 = b_in[n0 + 16 + l15];
    #pragma unroll
    for (int v = 0; v < 8; ++v) {
      const int r = msub * 16 + v + half * 8;
      zs[r * ZST + n0 + l15]      = acc0[v] + bias0;
      zs[r * ZST + n0 + 16 + l15] = acc1[v] + bias1;
    }
  }
  __syncthreads();

  // ---------- phase 2: argmin_k ( ||c_k||^2 - 2 z.c_k ) ----------
  {
    const int msub  = wave & 1;
    const int wslot = wave >> 1;                // 0..7
    float bestv[8];
    int   besti[8];
    #pragma unroll
    for (int v = 0; v < 8; ++v) { bestv[v] = 3.4e38f; besti[v] = 0; }

    const float* zr = zs + (msub * 16 + l15) * ZST + kb;
    // 16 iterations; each handles one chunk from each codebook half,
    // sharing the z fragment between two WMMA accumulator chains.
    for (int j = 0; j < 16; ++j) {
      const int n0 = (wslot + 8 * j) << 4;      // codes [0, 2048)
      const int n1 = n0 + (K_CB / 2);           // codes [2048, 4096)
      v8f acc0 = {}, acc1 = {};
      const float* cr0 = cb + (size_t)(n0 + l15) * D_EMB + kb;
      const float* cr1 = cb + (size_t)(n1 + l15) * D_EMB + kb;
      #pragma unroll 4
      for (int kk = 0; kk < D_EMB; kk += 4) {
        v2f a  = *(const v2f*)(zr + kk);
        v2f b0 = *(const v2f*)(cr0 + kk);
        v2f b1 = *(const v2f*)(cr1 + kk);
        acc0 = WMMA_F32(a, b0, acc0);
        acc1 = WMMA_F32(a, b1, acc1);
      }
      const float cn0 = cnorm[n0 + l15];
      const float cn1 = cnorm[n1 + l15];
      const int   c0  = n0 + l15;
      const int   c1  = n1 + l15;
      #pragma unroll
      for (int v = 0; v < 8; ++v) {
        const float s0 = cn0 - 2.0f * acc0[v];   // compare smaller index first
        if (s0 < bestv[v] || (s0 == bestv[v] && c0 < besti[v])) { bestv[v] = s0; besti[v] = c0; }
        const float s1 = cn1 - 2.0f * acc1[v];
        if (s1 < bestv[v] || (s1 == bestv[v] && c1 < besti[v])) { bestv[v] = s1; besti[v] = c1; }
      }
    }
    // min/argmin across the 16 lanes that share a token row (xor<16 keeps halves apart)
    #pragma unroll
    for (int m = 1; m < 16; m <<= 1) {
      #pragma unroll
      for (int v = 0; v < 8; ++v) {
        const float ov = __shfl_xor(bestv[v], m, 32);
        const int   oi = __shfl_xor(besti[v], m, 32);
        if (ov < bestv[v] || (ov == bestv[v] && oi < besti[v])) { bestv[v] = ov; besti[v] = oi; }
      }
    }
    if (l15 == 0) {
      #pragma unroll
      for (int v = 0; v < 8; ++v) {
        const int r = msub * 16 + v + half * 8;
        minv[r * 8 + wslot] = bestv[v];
        mini[r * 8 + wslot] = besti[v];
      }
    }
  }
  __syncthreads();

  if (tid < TOKT) {        // final cross-wave reduce, first-index tie break
    float bv = minv[tid * 8];
    int   bi = mini[tid * 8];
    #pragma unroll
    for (int w = 1; w < 8; ++w) {
      const float ov = minv[tid * 8 + w];
      const int   oi = mini[tid * 8 + w];
      if (ov < bv || (ov == bv && oi < bi)) { bv = ov; bi = oi; }
    }
    idxs[tid] = bi;
    outf[IDX_OFF + (size_t)(tok0 + tid)] = (float)bi;
  }
  __syncthreads();

  // ---------- phase 3: gather q (== quantized), aux partial ----------
  float part = 0.f;
  for (int i = tid; i < TOKT * D_EMB; i += THREADS) {
    const int t = i >> 10, e = i & (D_EMB - 1);
    const float qv = cb[(size_t)idxs[t] * D_EMB + e];
    const float zv = zs[t * ZST + e];
    const float d  = qv - zv;
    part += d * d;
    qs[t * ZST + e] = qv;
    outf[Q_OFF + (size_t)(tok0 + t) * D_EMB + e] = qv;
  }
  red[tid] = part;
  __syncthreads();                              // also guards qs before phase 4
  for (int s = THREADS / 2; s > 0; s >>= 1) {
    if (tid < s) red[tid] += red[tid + s];
    __syncthreads();
  }
  if (tid == 0) partials[blockIdx.x] = red[0];

  // ---------- phase 4: out = q @ w_out^T + b_out ----------
  // 2 token-subtiles x 16 column-pairs = 32 units; A fragment shared by the pair.
  for (int unit = wave; unit < 32; unit += NWAVES) {
    const int msub = unit >> 4;
    const int n0   = (unit & 15) << 5;
    v8f acc0 = {}, acc1 = {};
    const float* qr  = qs    + (msub * 16 + l15) * ZST + kb;
    const float* wr0 = w_out + (size_t)(n0 + l15) * D_EMB + kb;
    const float* wr1 = wr0   + (size_t)16 * D_EMB;
    #pragma unroll 4
    for (int kk = 0; kk < D_EMB; kk += 4) {
      v2f a  = *(const v2f*)(qr + kk);
      v2f b0 = *(const v2f*)(wr0 + kk);
      v2f b1 = *(const v2f*)(wr1 + kk);
      acc0 = WMMA_F32(a, b0, acc0);
      acc1 = WMMA_F32(a, b1, acc1);
    }
    const float bias0 = b_out[n0 + l15];
    const float bias1 = b_out[n0 + 16 + l15];
    #pragma unroll
    for (int v = 0; v < 8; ++v) {
      const int r = msub * 16 + v + half * 8;
      outf[OUT_OFF + (size_t)(tok0 + r) * D_IN + n0 + l15]      = acc0[v] + bias0;
      outf[OUT_OFF + (size_t)(tok0 + r) * D_IN + n0 + 16 + l15] = acc1[v] + bias1;
    }
  }
}

// ---------------- kernel 2: deterministic aux_loss finalize ----------------
__global__ __launch_bounds__(256) void vq_aux_kernel(const float* __restrict__ partials,
                                                     float* __restrict__ outf) {
  __shared__ float red[256];
  float s = 0.f;
  for (int i = threadIdx.x; i < NBLOCKS; i += 256) s += partials[i];
  red[threadIdx.x] = s;
  __syncthreads();
  for (int st = 128; st > 0; st >>= 1) {
    if (threadIdx.x < st) red[threadIdx.x] += red[threadIdx.x + st];
    __syncthreads();
  }
  if (threadIdx.x == 0)
    outf[AUX_OFF] = red[0] * (1.0f / ((float)T_TOT * (float)D_EMB));
}

extern "C" void kernel_launch(void* const* d_in, const int* in_sizes, int n_in,
                              void* d_out, int out_size, void* d_ws, size_t ws_size,
                              hipStream_t stream) {
  const float* x     = (const float*)d_in[0];
  const float* w_in  = (const float*)d_in[1];
  const float* b_in  = (const float*)d_in[2];
  const float* w_out = (const float*)d_in[3];
  const float* b_out = (const float*)d_in[4];
  const float* cb    = (const float*)d_in[5];
  float* outf        = (float*)d_out;
  float* ws          = (float*)d_ws;
  float* cnorm       = ws + WS_CNORM;
  float* partials    = ws + WS_PARTIALS;

  vq_cnorm_kernel<<<K_CB / 8, 256, 0, stream>>>(cb, cnorm);
  vq_fused_kernel<<<NBLOCKS, THREADS, LDS_BYTES, stream>>>(
      x, w_in, b_in, w_out, b_out, cb, cnorm, partials, outf);
  vq_aux_kernel<<<1, 256, 0, stream>>>(partials, outf);
}